// LightweightMemory_19490561589568
// MI455X (gfx1250) — compile-verified
//
#include <hip/hip_runtime.h>
#include <hip/hip_bf16.h>
#include <math.h>

#define B_    32
#define S_    128
#define D_    512
#define N_    131072
#define R_    16
#define K_TOP 8

// Async global->LDS staging (CDNA5 GLOBAL_LOAD_ASYNC_TO_LDS_B128 + ASYNCcnt).
// Set to 0 to fall back to synchronous VGPR staging.
#define ASYNC_STAGE 1

typedef float v2f __attribute__((ext_vector_type(2)));
typedef float v4f __attribute__((ext_vector_type(4)));
typedef float v8f __attribute__((ext_vector_type(8)));

#if ASYNC_STAGE
// copies 16 bytes per lane from the global address g to the LDS address l, tracked by ASYNCcnt.
__device__ __forceinline__ void async_b128(const float* g, const float* l) {
    unsigned loff = (unsigned)(uintptr_t)l;   // DS addressing uses addr[31:0]
    asm volatile("global_load_async_to_lds_b128 %0, %1, off"
                 :: "v"(loff), "v"(g) : "memory");
}
__device__ __forceinline__ void wait_async0() {
    asm volatile("s_wait_asynccnt 0x0" ::: "memory");
}
#endif

// ---------------------------------------------------------------------------
// q[b,d] = mean_s query[b,s,d]
__global__ void k_mean_q(const float* __restrict__ query, float* __restrict__ q) {
    int i = blockIdx.x * 256 + threadIdx.x;          // 0 .. B*D-1
    if (i >= B_ * D_) return;
    int b = i >> 9, d = i & (D_ - 1);
    const float* p = query + (size_t)b * S_ * D_ + d;
    float s = 0.f;
#pragma unroll 4
    for (int t = 0; t < S_; ++t) s += p[(size_t)t * D_];
    q[i] = s * (1.0f / S_);
}

// t[b,r] = lora_B[r,:] . q[b,:]
__global__ void k_lora_t(const float* __restrict__ lB, const float* __restrict__ q,
                         float* __restrict__ t) {
    int i = blockIdx.x * 256 + threadIdx.x;          // 0 .. B*R-1
    if (i >= B_ * R_) return;
    int b = i >> 4, r = i & 15;
    float s = 0.f;
    for (int d = 0; d < D_; ++d) s += lB[r * D_ + d] * q[b * D_ + d];
    t[i] = s;
}

// ---------------------------------------------------------------------------
// out[m,n] = sum_k X[m,k]*W[n,k]  (+ sum_r Xl[m,r]*Wl[n,r] if LORA) (+ bias[n])
// M = 32 fixed, K = 512 fixed, Ntot multiple of 64.
// 256 threads = 8 waves; wave (mt, ct) owns a 16x16 fp32 WMMA tile.
// Double-buffered LDS K-chunks (128 wide), staged with async-to-LDS copies.
// Row stride 132 floats: 16B-aligned rows for B128 LDS writes, banks spread 4*dn.
template <bool LORA>
__global__ __launch_bounds__(256) void k_wmma_gemm_nt(
    const float* __restrict__ X,    // [32, 512]
    const float* __restrict__ W,    // [Ntot, 512]
    const float* __restrict__ Xl,   // [32, 16]   (LORA)
    const float* __restrict__ Wl,   // [Ntot, 16] (LORA)
    const float* __restrict__ bias, // [Ntot] or nullptr
    float* __restrict__ out,        // [32, Ntot]
    int Ntot) {
    __shared__ float ldsW[2][64 * 132];   // ~66 KB
    __shared__ float ldsX[2][32 * 132];   // ~33 KB

    const int tid  = threadIdx.x;
    const int lane = tid & 31;
    const int wave = tid >> 5;
    const int lm   = lane & 15;        // M index (A) / N index (B,C)
    const int hi   = lane >> 4;        // K-half select (A,B) / M+8 select (C)
    const int n0   = blockIdx.x * 64;
    const int mt   = (wave >> 2) * 16; // 0 or 16
    const int ct   = (wave & 3) * 16;  // column sub-tile

    auto stage = [&](int buf, int kc) {
#if ASYNC_STAGE
#pragma unroll
        for (int j = 0; j < 8; ++j) {                 // 2048 b128 for W tile
            int idx = tid + j * 256;
            int row = idx >> 5, c4 = (idx & 31) << 2;
            async_b128(W + (size_t)(n0 + row) * D_ + kc * 128 + c4,
                       &ldsW[buf][row * 132 + c4]);
        }
#pragma unroll
        for (int j = 0; j < 4; ++j) {                 // 1024 b128 for X tile
            int idx = tid + j * 256;
            int row = idx >> 5, c4 = (idx & 31) << 2;
            async_b128(X + (size_t)row * D_ + kc * 128 + c4,
                       &ldsX[buf][row * 132 + c4]);
        }
#else
#pragma unroll
        for (int j = 0; j < 8; ++j) {
            int idx = tid + j * 256;
            int row = idx >> 5, c4 = (idx & 31) << 2;
            const v4f v = *(const v4f*)(W + (size_t)(n0 + row) * D_ + kc * 128 + c4);
            float* dst = &ldsW[buf][row * 132 + c4];
            dst[0] = v.x; dst[1] = v.y; dst[2] = v.z; dst[3] = v.w;
        }
#pragma unroll
        for (int j = 0; j < 4; ++j) {
            int idx = tid + j * 256;
            int row = idx >> 5, c4 = (idx & 31) << 2;
            const v4f v = *(const v4f*)(X + (size_t)row * D_ + kc * 128 + c4);
            float* dst = &ldsX[buf][row * 132 + c4];
            dst[0] = v.x; dst[1] = v.y; dst[2] = v.z; dst[3] = v.w;
        }
#endif
    };
#if ASYNC_STAGE
#define STAGE_FENCE() do { wait_async0(); __syncthreads(); } while (0)
#else
#define STAGE_FENCE() __syncthreads()
#endif

    stage(0, 0);                       // prefetch first K-chunk
    STAGE_FENCE();

    v8f c = {};
    for (int kc = 0; kc < 4; ++kc) {
        const int cur = kc & 1;
        if (kc < 3) stage(1 - cur, kc + 1);   // async prefetch next chunk
        const float* lw = &ldsW[cur][(ct + lm) * 132];
        const float* lx = &ldsX[cur][(mt + lm) * 132];
#pragma unroll
        for (int k = 0; k < 128; k += 4) {
            v2f a, b;
            a.x = lx[k + hi * 2]; a.y = lx[k + hi * 2 + 1];
            b.x = lw[k + hi * 2]; b.y = lw[k + hi * 2 + 1];
            c = __builtin_amdgcn_wmma_f32_16x16x4_f32(false, a, false, b,
                                                      (short)0, c, false, false);
        }
        STAGE_FENCE();                 // my prefetch landed + all waves done reading
    }

    if constexpr (LORA) {
        // Fused LoRA contribution: extra K=16 slab (t and lora_A, direct from L2)
#pragma unroll
        for (int kk = 0; kk < 4; ++kk) {
            v2f a, b;
            const float* xa = Xl + (mt + lm) * R_ + kk * 4 + hi * 2;
            a.x = xa[0]; a.y = xa[1];
            const float* wb = Wl + (size_t)(n0 + ct + lm) * R_ + kk * 4 + hi * 2;
            b.x = wb[0]; b.y = wb[1];
            c = __builtin_amdgcn_wmma_f32_16x16x4_f32(false, a, false, b,
                                                      (short)0, c, false, false);
        }
    }
    // C layout: lanes 0-15 -> N=lane, M=v ; lanes 16-31 -> N=lane-16, M=v+8
    const int ncol = n0 + ct + lm;
    const float bv = bias ? bias[ncol] : 0.0f;
#pragma unroll
    for (int v = 0; v < 8; ++v) {
        int m = mt + v + hi * 8;
        out[(size_t)m * Ntot + ncol] = c[v] + bv;
    }
#undef STAGE_FENCE
}

// ---------------------------------------------------------------------------
// 8 sequential block-wide argmax passes per batch row (row lives in L2).
__global__ void k_topk(float* __restrict__ scores, int* __restrict__ top_idx) {
    __shared__ float sv[256];
    __shared__ int   si[256];
    const int b = blockIdx.x;
    float* row = scores + (size_t)b * N_;
    for (int k = 0; k < K_TOP; ++k) {
        float best = -INFINITY; int bidx = N_;
        for (int i = threadIdx.x; i < N_; i += 256) {
            float v = row[i];
            if (v > best || (v == best && i < bidx)) { best = v; bidx = i; }
        }
        sv[threadIdx.x] = best; si[threadIdx.x] = bidx;
        __syncthreads();
        for (int s = 128; s > 0; s >>= 1) {
            if (threadIdx.x < s) {
                float ov = sv[threadIdx.x + s]; int oi = si[threadIdx.x + s];
                if (ov > sv[threadIdx.x] ||
                    (ov == sv[threadIdx.x] && oi < si[threadIdx.x])) {
                    sv[threadIdx.x] = ov; si[threadIdx.x] = oi;
                }
            }
            __syncthreads();
        }
        if (threadIdx.x == 0) {
            top_idx[b * K_TOP + k] = si[0];
            row[si[0]] = -INFINITY;           // exclude for next pass
        }
        __syncthreads();
    }
}

// retrieved[bk,:] = base[idx] + lora_A[idx] @ lora_B    (256 rows total)
__global__ void k_retrieved(const float* __restrict__ base, const float* __restrict__ lA,
                            const float* __restrict__ lB, const int* __restrict__ top_idx,
                            float* __restrict__ out_ret) {
    const int bk  = blockIdx.x;                // 0..255
    const int idx = top_idx[bk];
    const int d   = threadIdx.x * 4;           // 128 threads x float4
    v4f acc = *(const v4f*)(base + (size_t)idx * D_ + d);
    const float* a = lA + (size_t)idx * R_;
#pragma unroll
    for (int r = 0; r < R_; ++r) {
        float ar = a[r];
        const v4f bv = *(const v4f*)(lB + (size_t)r * D_ + d);
        acc.x += ar * bv.x; acc.y += ar * bv.y; acc.z += ar * bv.z; acc.w += ar * bv.w;
    }
    *(v4f*)(out_ret + (size_t)bk * D_ + d) = acc;
}

__global__ void k_xsum(const float* __restrict__ ret, float* __restrict__ xsum) {
    int b = blockIdx.x;
    for (int d = threadIdx.x; d < D_; d += 256) {
        float s = 0.f;
#pragma unroll
        for (int k = 0; k < K_TOP; ++k) s += ret[((size_t)b * K_TOP + k) * D_ + d];
        xsum[b * D_ + d] = s;
    }
}

// h0 == 0  =>  gh = b_hh,  hidden = (1-z)*n
__global__ void k_gru(const float* __restrict__ gx, const float* __restrict__ b_hh,
                      float* __restrict__ hidden) {
    int i = blockIdx.x * 256 + threadIdx.x;
    if (i >= B_ * D_) return;
    int b = i >> 9, d = i & (D_ - 1);
    const float* g = gx + (size_t)b * 3 * D_;
    float r = 1.f / (1.f + expf(-(g[d]          + b_hh[d])));
    float z = 1.f / (1.f + expf(-(g[D_ + d]     + b_hh[D_ + d])));
    float n = tanhf(g[2 * D_ + d] + r * b_hh[2 * D_ + d]);
    hidden[i] = (1.f - z) * n;
}

__global__ void k_wprob(const float* __restrict__ hidden, const float* __restrict__ wg_w,
                        const float* __restrict__ wg_b, float* __restrict__ wprob) {
    __shared__ float sv[256];
    int b = blockIdx.x;
    float s = 0.f;
    for (int d = threadIdx.x; d < D_; d += 256) s += hidden[b * D_ + d] * wg_w[d];
    sv[threadIdx.x] = s; __syncthreads();
    for (int st = 128; st > 0; st >>= 1) {
        if (threadIdx.x < st) sv[threadIdx.x] += sv[threadIdx.x + st];
        __syncthreads();
    }
    if (threadIdx.x == 0) wprob[b] = 1.f / (1.f + expf(-(sv[0] + wg_b[0])));
}

// Streaming 256 MB copy base -> new_mem; non-temporal so L2 keeps `scores`.
__global__ void k_copy(const v4f* __restrict__ src, v4f* __restrict__ dst, long n4) {
    long i = (long)blockIdx.x * blockDim.x + threadIdx.x;
    long stride = (long)gridDim.x * blockDim.x;
    for (; i < n4; i += stride) {
        v4f v = __builtin_nontemporal_load(&src[i]);
        __builtin_nontemporal_store(v, &dst[i]);
    }
}

// Sequential-over-batch scatter (matches lax.scan order). Single block;
// within a batch the 8 indices are distinct -> parallel update is exact.
__global__ void k_scatter(float* __restrict__ newmem, const int* __restrict__ top_idx,
                          const float* __restrict__ wprob, const float* __restrict__ q) {
    for (int b = 0; b < B_; ++b) {
        float p = wprob[b];
        for (int i = threadIdx.x; i < K_TOP * D_; i += 256) {
            int k = i >> 9, d = i & (D_ - 1);
            int idx = top_idx[b * K_TOP + k];
            size_t off = (size_t)idx * D_ + d;
            newmem[off] = (1.f - p) * newmem[off] + p * q[b * D_ + d];
        }
        __syncthreads();   // cross-batch RAW (repeated indices across batches)
    }
}

// ---------------------------------------------------------------------------
extern "C" void kernel_launch(void* const* d_in, const int* in_sizes, int n_in,
                              void* d_out, int out_size, void* d_ws, size_t ws_size,
                              hipStream_t stream) {
    (void)in_sizes; (void)n_in; (void)out_size; (void)ws_size;

    const float* query = (const float*)d_in[0];
    const float* base  = (const float*)d_in[1];
    const float* lA    = (const float*)d_in[2];
    const float* lB    = (const float*)d_in[3];
    const float* w_ih  = (const float*)d_in[4];
    /* d_in[5] = gru_w_hh unused: h0 == 0 */
    const float* b_ih  = (const float*)d_in[6];
    const float* b_hh  = (const float*)d_in[7];
    const float* wg_w  = (const float*)d_in[8];
    const float* wg_b  = (const float*)d_in[9];

    float* ws     = (float*)d_ws;
    float* q      = ws + 0;        // 16384
    float* t      = ws + 16384;    // 512
    float* xsum   = ws + 16896;    // 16384
    float* gx     = ws + 33280;    // 49152
    float* wprob  = ws + 82432;    // 32
    int*   tidx   = (int*)(ws + 82464); // 256 ints
    float* scores = ws + 83968;    // 4194304  (~17 MB total ws use)

    float* out_ret = (float*)d_out;                 // [B,K,D]
    float* out_hid = out_ret + B_ * K_TOP * D_;     // [B,D]
    float* out_mem = out_hid + B_ * D_;             // [N,D]

    k_mean_q<<<(B_ * D_ + 255) / 256, 256, 0, stream>>>(query, q);
    k_lora_t<<<2, 256, 0, stream>>>(lB, q, t);
    k_wmma_gemm_nt<true><<<N_ / 64, 256, 0, stream>>>(q, base, t, lA, nullptr, scores, N_);
    k_topk<<<B_, 256, 0, stream>>>(scores, tidx);
    k_retrieved<<<B_ * K_TOP, 128, 0, stream>>>(base, lA, lB, tidx, out_ret);
    k_xsum<<<B_, 256, 0, stream>>>(out_ret, xsum);
    k_wmma_gemm_nt<false><<<(3 * D_) / 64, 256, 0, stream>>>(xsum, w_ih, nullptr, nullptr,
                                                             b_ih, gx, 3 * D_);
    k_gru<<<(B_ * D_ + 255) / 256, 256, 0, stream>>>(gx, b_hh, out_hid);
    k_wprob<<<B_, 256, 0, stream>>>(out_hid, wg_w, wg_b, wprob);
    k_copy<<<4096, 256, 0, stream>>>((const v4f*)base, (v4f*)out_mem,
                                     (long)N_ * D_ / 4);
    k_scatter<<<1, 256, 0, stream>>>(out_mem, tidx, wprob, q);
}